// GCNNetLayer_19095424598405
// MI455X (gfx1250) — compile-verified
//
#include <hip/hip_runtime.h>
#include <math.h>

#define N_NODES 100000
#define E_EDGES 3200000
#define DIM     256
#define LN_EPS  1e-5f

typedef __attribute__((ext_vector_type(2))) float v2f;
typedef __attribute__((ext_vector_type(8))) float v8f;

__device__ __forceinline__ float safef(float v) {
    return __builtin_isfinite(v) ? v : 0.0f;
}

// ---------------------------------------------------------------- h = safe(x)
__global__ void k_copy_safe(const float* __restrict__ x, float* __restrict__ h, int n) {
    int i = blockIdx.x * blockDim.x + threadIdx.x;
    if (i < n) h[i] = safef(x[i]);
}

__global__ void k_zero(float* __restrict__ p, int n) {
    int i = blockIdx.x * blockDim.x + threadIdx.x;
    if (i < n) p[i] = 0.0f;
}

// ------------------------------------------------- deg[col] += clamped |ew|
__global__ void k_deg_scatter(const int* __restrict__ ei, const float* __restrict__ ew,
                              float* __restrict__ deg) {
    int e = blockIdx.x * blockDim.x + threadIdx.x;
    if (e >= E_EDGES) return;
    float w = fmaxf(fabsf(safef(ew[e])), 1e-6f);
    int col = ei[E_EDGES + e];
    atomicAdd(&deg[col], w);
}

// ---------------------------------------------- dis = rsqrt(deg+1), in place
__global__ void k_dis(float* __restrict__ d, int n) {
    int i = blockIdx.x * blockDim.x + threadIdx.x;
    if (i < n) d[i] = rsqrtf(d[i] + 1.0f);
}

// --------------------------------------------------------------- xw = h @ W
// fp32 WMMA 16x16x4. One wave = one 16x16 C tile; 4 waves / block.
// Tiles: (N/16)=6250 row tiles x (D/16)=16 col tiles = 100000 tiles.
// A layout (16x4 f32): lane l (l<16): VGPR0=A[l][k],   VGPR1=A[l][k+1]
//                      lane l+16   : VGPR0=A[l][k+2], VGPR1=A[l][k+3]
// B layout (4x16 f32): lane l: {B[k][c0+l], B[k+1][c0+l]}; upper half K+2.
// C/D layout: VGPR v -> row (v + 8*half), col l.
__global__ void __launch_bounds__(128) k_gemm_wmma(const float* __restrict__ A,
                                                   const float* __restrict__ W,
                                                   float* __restrict__ C) {
    const int wave = threadIdx.x >> 5;
    const int lane = threadIdx.x & 31;
    const int l    = lane & 15;
    const int half = lane >> 4;

    const int tile = blockIdx.x * 4 + wave;      // 0..99999
    const int tm   = tile >> 4;                  // 0..6249
    const int tn   = tile & 15;                  // 0..15
    const int r0   = tm * 16;
    const int c0   = tn * 16;

    const float* __restrict__ arow = A + (size_t)(r0 + l) * DIM; // this lane's A row
    const float* __restrict__ wcol = W + c0 + l;                 // this lane's B column

    v8f acc = {};
#pragma unroll 8
    for (int k = 0; k < DIM; k += 4) {
        const int ka = k + 2 * half;
        v2f a = *(const v2f*)(arow + ka);                    // 8B-aligned b64 load
        v2f b;
        b.x = wcol[(size_t)ka * DIM];
        b.y = wcol[(size_t)(ka + 1) * DIM];
        acc = __builtin_amdgcn_wmma_f32_16x16x4_f32(
            /*neg_a=*/false, a, /*neg_b=*/false, b,
            /*c_mod=*/(short)0, acc, /*reuse_a=*/false, /*reuse_b=*/false);
    }

    float* __restrict__ crow = C + (size_t)(r0 + 8 * half) * DIM + c0 + l;
#pragma unroll
    for (int v = 0; v < 8; ++v) crow[(size_t)v * DIM] = acc[v];
}

// ----------------------------------- agg = dis^2 * xw + bias  (self loop + b)
__global__ void k_agg_init(const float* __restrict__ xw, const float* __restrict__ dis,
                           const float* __restrict__ bias, float* __restrict__ agg) {
    int i = blockIdx.x * blockDim.x + threadIdx.x;   // < N*D = 25.6M
    int row = i >> 8;
    int j   = i & 255;
    float d = dis[row];
    agg[i] = d * d * xw[i] + bias[j];
}

// ---------------------- per-edge: agg[col] += dis[row]*ew*dis[col] * xw[row]
// one wave per edge; each lane handles 8 contiguous columns (2x float4 gather)
__global__ void __launch_bounds__(256) k_edge_scatter(const int* __restrict__ ei,
                                                      const float* __restrict__ ew,
                                                      const float* __restrict__ dis,
                                                      const float* __restrict__ xw,
                                                      float* __restrict__ agg) {
    const int wave = threadIdx.x >> 5;
    const int lane = threadIdx.x & 31;
    const int e = blockIdx.x * 8 + wave;             // E/8 = 400000 blocks, exact
    const int row = ei[e];
    const int col = ei[E_EDGES + e];
    const float w = fmaxf(fabsf(safef(ew[e])), 1e-6f);
    const float norm = dis[row] * w * dis[col];

    const float4* __restrict__ src = (const float4*)(xw + (size_t)row * DIM) + lane * 2;
    float4 p = src[0];
    float4 q = src[1];
    float* __restrict__ dst = agg + (size_t)col * DIM + lane * 8;
    atomicAdd(dst + 0, norm * p.x);
    atomicAdd(dst + 1, norm * p.y);
    atomicAdd(dst + 2, norm * p.z);
    atomicAdd(dst + 3, norm * p.w);
    atomicAdd(dst + 4, norm * q.x);
    atomicAdd(dst + 5, norm * q.y);
    atomicAdd(dst + 6, norm * q.z);
    atomicAdd(dst + 7, norm * q.w);
}

// -------- h = safe( relu( safe(LN(safe(agg))) ) + h )   (one wave32 per row)
__global__ void __launch_bounds__(256) k_ln_relu_res(const float* __restrict__ agg,
                                                     const float* __restrict__ g,
                                                     const float* __restrict__ be,
                                                     float* __restrict__ h) {
    const int wave = threadIdx.x >> 5;
    const int lane = threadIdx.x & 31;
    const int row = blockIdx.x * 8 + wave;           // N/8 = 12500 blocks, exact

    const float4* __restrict__ arow = (const float4*)(agg + (size_t)row * DIM) + lane * 2;
    float4 p = arow[0];
    float4 q = arow[1];
    float vals[8] = { safef(p.x), safef(p.y), safef(p.z), safef(p.w),
                      safef(q.x), safef(q.y), safef(q.z), safef(q.w) };

    float s = 0.0f, ss = 0.0f;
#pragma unroll
    for (int t = 0; t < 8; ++t) { s += vals[t]; ss += vals[t] * vals[t]; }
#pragma unroll
    for (int off = 16; off >= 1; off >>= 1) {
        s  += __shfl_xor(s,  off, 32);
        ss += __shfl_xor(ss, off, 32);
    }
    const float mu  = s * (1.0f / DIM);
    const float var = fmaxf(ss * (1.0f / DIM) - mu * mu, 0.0f);
    const float inv = rsqrtf(var + LN_EPS);

    const int j0 = lane * 8;
    float* __restrict__ hrow = h + (size_t)row * DIM + j0;
    float out[8];
#pragma unroll
    for (int t = 0; t < 8; ++t) {
        float v = safef((vals[t] - mu) * inv * g[j0 + t] + be[j0 + t]);
        v = fmaxf(v, 0.0f);                    // relu
        out[t] = safef(v + hrow[t]);           // residual (read-before-write, same thread)
    }
    float4 o0 = { out[0], out[1], out[2], out[3] };
    float4 o1 = { out[4], out[5], out[6], out[7] };
    float4* __restrict__ hv = (float4*)hrow;
    hv[0] = o0;
    hv[1] = o1;
}

extern "C" void kernel_launch(void* const* d_in, const int* in_sizes, int n_in,
                              void* d_out, int out_size, void* d_ws, size_t ws_size,
                              hipStream_t stream) {
    const float* x  = (const float*)d_in[0];
    const int*   ei = (const int*)d_in[1];      // (2,E) int32 (JAX default x64 off)
    const float* ew = (const float*)d_in[2];
    const float* Wl[2]  = { (const float*)d_in[3], (const float*)d_in[7] };
    const float* bl[2]  = { (const float*)d_in[4], (const float*)d_in[8] };
    const float* gl[2]  = { (const float*)d_in[5], (const float*)d_in[9] };
    const float* bel[2] = { (const float*)d_in[6], (const float*)d_in[10] };

    float* h = (float*)d_out;                                  // [N, D] running features

    const size_t XW_BYTES = (size_t)N_NODES * DIM * sizeof(float);   // 102.4 MB
    char* ws   = (char*)d_ws;
    float* xw  = (float*)ws;                                   // [N, D]
    float* agg = (float*)(ws + XW_BYTES);                      // [N, D]
    float* deg = (float*)(ws + 2 * XW_BYTES);                  // [N], becomes dis in place

    const int ND = N_NODES * DIM;

    // h = safe(x); deg = 0
    k_copy_safe<<<(ND + 255) / 256, 256, 0, stream>>>(x, h, ND);
    k_zero<<<(N_NODES + 255) / 256, 256, 0, stream>>>(deg, N_NODES);
    // deg scatter + rsqrt
    k_deg_scatter<<<(E_EDGES + 255) / 256, 256, 0, stream>>>(ei, ew, deg);
    k_dis<<<(N_NODES + 255) / 256, 256, 0, stream>>>(deg, N_NODES);
    const float* dis = deg;

    for (int layer = 0; layer < 2; ++layer) {
        // xw = h @ W   (WMMA fp32), 100000 tiles / 4 waves per block
        k_gemm_wmma<<<25000, 128, 0, stream>>>(h, Wl[layer], xw);
        // agg = dis^2 * xw + b
        k_agg_init<<<ND / 256, 256, 0, stream>>>(xw, dis, bl[layer], agg);
        // scatter-add messages
        k_edge_scatter<<<E_EDGES / 8, 256, 0, stream>>>(ei, ew, dis, xw, agg);
        // h = safe(relu(LN(agg)) + h)
        k_ln_relu_res<<<N_NODES / 8, 256, 0, stream>>>(agg, gl[layer], bel[layer], h);
    }
}